// ConvBlock_54125177864328
// MI455X (gfx1250) — compile-verified
//
#include <hip/hip_runtime.h>
#include <math.h>

// ---------------------------------------------------------------------------
// Problem constants (from reference): B=64, L=512, D0=300, K=64, FW=4
// ---------------------------------------------------------------------------
#define B_   64
#define L_   512
#define D0_  300
#define K_   64
#define FW_  4
#define LP_  515          // L + FW - 1
#define LT_  33           // ceil(LP/16)
#define LTP_ (LT_ * 16)   // 528 padded rows for c0/c1

typedef __attribute__((ext_vector_type(2))) float v2f;
typedef __attribute__((ext_vector_type(4))) float v4f;
typedef __attribute__((ext_vector_type(8))) float v8f;
typedef __attribute__((ext_vector_type(4))) int   v4i;

// fp32 matrix pipe: D(16x16) += A(16x4) * B(4x16)
__device__ __forceinline__ v8f wmma4(v2f a, v2f b, v8f c) {
    return __builtin_amdgcn_wmma_f32_16x16x4_f32(
        /*neg_a=*/false, a, /*neg_b=*/false, b,
        /*c_mod=*/(short)0, c, /*reuse_a=*/false, /*reuse_b=*/false);
}

// ---------------------------------------------------------------------------
// Async global -> LDS staging (CDNA5 GLOBAL_LOAD_ASYNC_TO_LDS_B128 path,
// tracked with ASYNCcnt). Builtin expects int4-typed pointers; pass
// AS(1)/AS(3)-qualified ones (implicitly convertible to generic if needed).
// ---------------------------------------------------------------------------
#if defined(__gfx1250__) && __has_builtin(__builtin_amdgcn_global_load_async_to_lds_b128)
#define HAVE_ASYNC_LDS 1
typedef __attribute__((address_space(1))) v4i gbl_v4i;
typedef __attribute__((address_space(3))) v4i lds_v4i;
__device__ __forceinline__ void async_cp16(float* lds, const float* g) {
    __builtin_amdgcn_global_load_async_to_lds_b128(
        (gbl_v4i*)g, (lds_v4i*)lds, /*offset=*/0, /*cpol=*/0);
}
#define ASYNC_JOIN() asm volatile("s_wait_asynccnt 0" ::: "memory")
#else
__device__ __forceinline__ void async_cp16(float* lds, const float* g) {
    *(v4f*)lds = *(const v4f*)g;
}
#define ASYNC_JOIN() ((void)0)
#endif

// ---------------------------------------------------------------------------
// Row squared-norms: one wave per row of an (rows x d) matrix.
// ---------------------------------------------------------------------------
__global__ __launch_bounds__(256)
void row_norm_kernel(const float* __restrict__ x, float* __restrict__ n,
                     int rows, int d) {
    int row = blockIdx.x * (blockDim.x >> 5) + (threadIdx.x >> 5);
    if (row >= rows) return;
    int lane = threadIdx.x & 31;
    const float* p = x + (size_t)row * d;
    float s = 0.f;
    for (int i = lane; i < d; i += 32) { float v = p[i]; s += v * v; }
    s += __shfl_xor(s, 16, 32);
    s += __shfl_xor(s, 8, 32);
    s += __shfl_xor(s, 4, 32);
    s += __shfl_xor(s, 2, 32);
    s += __shfl_xor(s, 1, 32);
    if (lane == 0) n[row] = s;
}

// ---------------------------------------------------------------------------
// A[b,i,j] = 1/(1 + sqrt(max(|p_i|^2 + |q_j|^2 - 2 p_i.q_j, 0)))
// NT gram GEMM via WMMA. grid(L/16, 4, B), block 256 (8 waves, wave = j-tile).
// Shared P tile (16 x D0 = 19.2 KB) staged once per block via async copies;
// per-wave Q fragments stream from global.
// ---------------------------------------------------------------------------
__global__ __launch_bounds__(256)
void attn_kernel(const float* __restrict__ P, const float* __restrict__ Q,
                 const float* __restrict__ np_, const float* __restrict__ nq_,
                 float* __restrict__ A) {
    __shared__ float sp[16 * D0_];

    const int b    = blockIdx.z;
    const int wave = threadIdx.x >> 5;
    const int lane = threadIdx.x & 31;
    const int half = lane >> 4, l16 = lane & 15;
    const int i0 = blockIdx.x * 16;
    const int j0 = (blockIdx.y * 8 + wave) * 16;

    // cooperative stage: 16 rows x 300 f32 = 1200 x 16B chunks
    {
        const float* gsrc = P + ((size_t)b * L_ + i0) * D0_;
        const int nchunk = 16 * (D0_ / 4);
        for (int idx = threadIdx.x; idx < nchunk; idx += 256) {
            int row = idx / (D0_ / 4);
            int c4  = (idx % (D0_ / 4)) * 4;
            async_cp16(&sp[row * D0_ + c4], gsrc + (size_t)row * D0_ + c4);
        }
    }
    ASYNC_JOIN();
    __syncthreads();

    const float* qrow = Q + ((size_t)b * L_ + j0 + l16) * D0_;
    const float* arow = &sp[l16 * D0_];

    v8f acc = {};
    for (int k0 = 0; k0 < D0_; k0 += 4) {
        v2f a, bf;
        a.x  = arow[k0 + 2 * half];     a.y  = arow[k0 + 2 * half + 1];
        bf.x = qrow[k0 + 2 * half];     bf.y = qrow[k0 + 2 * half + 1];
        acc = wmma4(a, bf, acc);
    }

    const float qn = nq_[b * L_ + j0 + l16];   // column norm for this lane
    float* Ab = A + (size_t)b * L_ * L_;
#pragma unroll
    for (int m = 0; m < 8; ++m) {
        int i = i0 + half * 8 + m;
        float pn = np_[b * L_ + i];
        float sq = pn + qn - 2.f * acc[m];
        float d  = sqrtf(fmaxf(sq, 0.f));
        Ab[(size_t)i * L_ + j0 + l16] = 1.f / (1.f + d);
    }
}

// ---------------------------------------------------------------------------
// out = A @ W (TRANS=false) or A^T @ W (TRANS=true).  A: (L x L), W: (L x D0).
// grid(4, 19, B), block 256; wave handles i-tile = blockIdx.x*8 + wave.
// Shared W tile (512 x 16 = 32 KB, zero-padded past D0) staged once per block.
// ---------------------------------------------------------------------------
template <bool TRANS>
__global__ __launch_bounds__(256)
void aw_kernel(const float* __restrict__ A, const float* __restrict__ W,
               float* __restrict__ out) {
    __shared__ float sw[L_ * 16];   // [k][jj]

    const int b    = blockIdx.z;
    const int wave = threadIdx.x >> 5;
    const int lane = threadIdx.x & 31;
    const int half = lane >> 4, l16 = lane & 15;
    const int i0 = (blockIdx.x * 8 + wave) * 16;
    const int j0 = blockIdx.y * 16;

    // cooperative stage: 512 rows x 16 cols = 2048 x 16B chunks
    for (int idx = threadIdx.x; idx < L_ * 4; idx += 256) {
        int k  = idx >> 2;
        int c4 = (idx & 3) * 4;
        int j  = j0 + c4;
        if (j + 3 < D0_) {
            async_cp16(&sw[k * 16 + c4], &W[(size_t)k * D0_ + j]);
        } else {
#pragma unroll
            for (int e = 0; e < 4; ++e)
                sw[k * 16 + c4 + e] = (j + e < D0_) ? W[(size_t)k * D0_ + j + e] : 0.f;
        }
    }
    ASYNC_JOIN();
    __syncthreads();

    const float* Ab = A + (size_t)b * L_ * L_;
    v8f acc = {};
    for (int k0 = 0; k0 < L_; k0 += 4) {
        v2f a, bf;
        if (!TRANS) {
            const float* ar = Ab + (size_t)(i0 + l16) * L_ + k0 + 2 * half;
            a.x = ar[0]; a.y = ar[1];
        } else {
            a.x = Ab[(size_t)(k0 + 2 * half)     * L_ + i0 + l16];
            a.y = Ab[(size_t)(k0 + 2 * half + 1) * L_ + i0 + l16];
        }
        bf.x = sw[(k0 + 2 * half)     * 16 + l16];
        bf.y = sw[(k0 + 2 * half + 1) * 16 + l16];
        acc = wmma4(a, bf, acc);
    }

    if ((j0 + l16) < D0_) {
        float* ob = out + (size_t)b * L_ * D0_;
#pragma unroll
        for (int m = 0; m < 8; ++m)
            ob[(size_t)(i0 + half * 8 + m) * D0_ + j0 + l16] = acc[m];
    }
}

// ---------------------------------------------------------------------------
// Conv stage 1:  M[fh][b,l,k] = sum_{c,d} f[b,c,l,d] * w[k,c,fh,d]
// One A-fragment (f row data) feeds FW=4 WMMAs (one per filter tap).
// grid(4, K/16, B), block 256; wave handles l-tile = blockIdx.x*8 + wave.
// Weight array (600 KB) is grid-wide shared -> stays hot in the 192 MB L2.
// ---------------------------------------------------------------------------
__global__ __launch_bounds__(256)
void convm_kernel(const float* __restrict__ X,   // channel 0 (x part)
                  const float* __restrict__ Aa,  // channel 1 (attention part)
                  const float* __restrict__ Wc,  // (K, 2, FW, D0)
                  float* __restrict__ M) {       // (FW, B, L, K)
    const int b    = blockIdx.z;
    const int wave = threadIdx.x >> 5;
    const int lane = threadIdx.x & 31;
    const int half = lane >> 4, l16 = lane & 15;
    const int i0 = (blockIdx.x * 8 + wave) * 16;   // l tile
    const int j0 = blockIdx.y * 16;                // output channel tile
    const int kout = j0 + l16;

    v8f acc[FW_] = {};
    for (int c = 0; c < 2; ++c) {
        const float* src =
            (c == 0 ? X : Aa) + ((size_t)b * L_ + i0 + l16) * D0_;
        const float* wbase = Wc + ((size_t)kout * 2 + c) * FW_ * D0_;
        for (int k0 = 0; k0 < D0_; k0 += 4) {
            v2f a;
            a.x = src[k0 + 2 * half];
            a.y = src[k0 + 2 * half + 1];
#pragma unroll
            for (int fh = 0; fh < FW_; ++fh) {
                v2f bf;
                bf.x = wbase[fh * D0_ + k0 + 2 * half];
                bf.y = wbase[fh * D0_ + k0 + 2 * half + 1];
                acc[fh] = wmma4(a, bf, acc[fh]);
            }
        }
    }
#pragma unroll
    for (int fh = 0; fh < FW_; ++fh)
#pragma unroll
        for (int m = 0; m < 8; ++m) {
            int i = i0 + half * 8 + m;
            M[(((size_t)fh * B_ + b) * L_ + i) * K_ + j0 + l16] = acc[fh][m];
        }
}

// ---------------------------------------------------------------------------
// Conv stage 2: c[b,t,k] = tanh(bias[k] + sum_fh M[fh][b, t-3+fh, k]); zero pad
// rows t in [LP, LTP) so downstream WMMA tiles read zeros.
// ---------------------------------------------------------------------------
__global__ __launch_bounds__(256)
void tanhcomb_kernel(const float* __restrict__ M, const float* __restrict__ bias,
                     float* __restrict__ C) {
    int idx = blockIdx.x * blockDim.x + threadIdx.x;
    const int total = B_ * LTP_ * K_;
    if (idx >= total) return;
    int k = idx % K_;
    int t = (idx / K_) % LTP_;
    int b = idx / (K_ * LTP_);
    if (t >= LP_) { C[idx] = 0.f; return; }
    float s = bias[k];
#pragma unroll
    for (int fh = 0; fh < FW_; ++fh) {
        int l = t - (FW_ - 1) + fh;
        if (l >= 0 && l < L_)
            s += M[(((size_t)fh * B_ + b) * L_ + l) * K_ + k];
    }
    C[idx] = tanhf(s);
}

// ---------------------------------------------------------------------------
// A2 tile kernel: gram over K=64, transform to 1/(1+dist), and emit per-tile
// row/col partial sums (no float atomics -> deterministic).
// grid(LT, 5, B), block 256; wave j-tile = blockIdx.y*8 + wave (guarded).
// ---------------------------------------------------------------------------
__global__ __launch_bounds__(256)
void a2sum_kernel(const float* __restrict__ C0, const float* __restrict__ C1,
                  const float* __restrict__ n0, const float* __restrict__ n1,
                  float* __restrict__ rowpart,   // (B, LT, LTP) sums over i-tile
                  float* __restrict__ colpart) { // (B, LT, LTP) sums over j-tile
    const int b    = blockIdx.z;
    const int wave = threadIdx.x >> 5;
    const int jt   = blockIdx.y * 8 + wave;
    if (jt >= LT_) return;
    const int lane = threadIdx.x & 31;
    const int half = lane >> 4, l16 = lane & 15;
    const int it = blockIdx.x;
    const int i0 = it * 16;
    const int j0 = jt * 16;

    const float* p = C0 + ((size_t)b * LTP_ + i0 + l16) * K_;
    const float* q = C1 + ((size_t)b * LTP_ + j0 + l16) * K_;

    v8f acc = {};
    for (int k0 = 0; k0 < K_; k0 += 4) {
        v2f a, bf;
        a.x  = p[k0 + 2 * half];  a.y  = p[k0 + 2 * half + 1];
        bf.x = q[k0 + 2 * half];  bf.y = q[k0 + 2 * half + 1];
        acc = wmma4(a, bf, acc);
    }

    const bool  jv = (j0 + l16) < LP_;
    const float qn = n1[b * LTP_ + j0 + l16];
    float vals[8];
    float rsum = 0.f;
#pragma unroll
    for (int m = 0; m < 8; ++m) {
        int   i  = i0 + half * 8 + m;
        float pn = n0[b * LTP_ + i];
        float sq = pn + qn - 2.f * acc[m];
        float dd = sqrtf(fmaxf(sq, 0.f));
        float v  = (jv && i < LP_) ? 1.f / (1.f + dd) : 0.f;
        vals[m] = v;
        rsum += v;
    }
    // column sum over all 16 rows of the tile: combine the two half-lanes
    rsum += __shfl_xor(rsum, 16, 32);
    if (half == 0 && jv)
        rowpart[((size_t)b * LT_ + it) * LTP_ + j0 + l16] = rsum;

    // per-row sums: reduce across the 16 lanes of each half-group
#pragma unroll
    for (int m = 0; m < 8; ++m) {
        float cs = vals[m];
        cs += __shfl_xor(cs, 1, 32);
        cs += __shfl_xor(cs, 2, 32);
        cs += __shfl_xor(cs, 4, 32);
        cs += __shfl_xor(cs, 8, 32);
        int i = i0 + half * 8 + m;
        if (l16 == 0 && i < LP_)
            colpart[((size_t)b * LT_ + jt) * LTP_ + i] = cs;
    }
}

// Fixed-order reduction of the LT_ partial planes -> row / col vectors.
__global__ __launch_bounds__(256)
void reduce_parts_kernel(const float* __restrict__ part, float* __restrict__ out) {
    int idx = blockIdx.x * blockDim.x + threadIdx.x;
    if (idx >= B_ * LP_) return;
    int b = idx / LP_, j = idx % LP_;
    float s = 0.f;
    for (int it = 0; it < LT_; ++it)
        s += part[((size_t)b * LT_ + it) * LTP_ + j];
    out[idx] = s;
}

// wp[b,t,k] = sum_{s=t..t+FW-1} wgt[b,s] * C[b,s,k]   (t in [0,L))
__global__ __launch_bounds__(256)
void wp_kernel(const float* __restrict__ C, const float* __restrict__ wgt,
               float* __restrict__ out) {
    int idx = blockIdx.x * blockDim.x + threadIdx.x;
    if (idx >= B_ * L_ * K_) return;
    int k = idx % K_;
    int t = (idx / K_) % L_;
    int b = idx / (K_ * L_);
    float s = 0.f;
#pragma unroll
    for (int fw = 0; fw < FW_; ++fw) {
        int ss = t + fw;   // < LP_
        s += wgt[b * LP_ + ss] * C[((size_t)b * LTP_ + ss) * K_ + k];
    }
    out[idx] = s;
}

// ap[b,k] = mean_t C[b,t,k]  (pad rows are zero; divide by LP)
__global__ __launch_bounds__(64)
void ap_kernel(const float* __restrict__ C, float* __restrict__ out) {
    int b = blockIdx.x;
    int k = threadIdx.x;
    float s = 0.f;
    for (int t = 0; t < LP_; ++t)
        s += C[((size_t)b * LTP_ + t) * K_ + k];
    out[b * K_ + k] = s * (1.0f / (float)LP_);
}

// ---------------------------------------------------------------------------
// Host-side orchestration
// ---------------------------------------------------------------------------
extern "C" void kernel_launch(void* const* d_in, const int* in_sizes, int n_in,
                              void* d_out, int out_size, void* d_ws, size_t ws_size,
                              hipStream_t stream) {
    (void)in_sizes; (void)n_in; (void)out_size;

    const float* x0 = (const float*)d_in[0];   // (B,1,L,D0)
    const float* x1 = (const float*)d_in[1];
    const float* W0 = (const float*)d_in[2];   // (L,D0)
    const float* W1 = (const float*)d_in[3];
    const float* cw = (const float*)d_in[4];   // (K,2,FW,D0)
    const float* cb = (const float*)d_in[5];   // (K,)
    float* out = (float*)d_out;

    // ---- workspace carve-out (floats) ----
    float* ws = (float*)d_ws;
    size_t off = 0;
    auto carve = [&](size_t n) { float* p = ws + off; off += n; return p; };
    float* A    = carve((size_t)B_ * L_ * L_);      // 16.78M
    float* a0   = carve((size_t)B_ * L_ * D0_);     //  9.83M
    float* a1   = carve((size_t)B_ * L_ * D0_);
    float* np_  = carve((size_t)B_ * L_);
    float* nq_  = carve((size_t)B_ * L_);
    float* M    = carve((size_t)FW_ * B_ * L_ * K_);
    float* c0   = carve((size_t)B_ * LTP_ * K_);
    float* c1   = carve((size_t)B_ * LTP_ * K_);
    float* nc0  = carve((size_t)B_ * LTP_);
    float* nc1  = carve((size_t)B_ * LTP_);
    float* rpart= carve((size_t)B_ * LT_ * LTP_);
    float* cpart= carve((size_t)B_ * LT_ * LTP_);
    float* rowv = carve((size_t)B_ * LP_);
    float* colv = carve((size_t)B_ * LP_);
    if (off * sizeof(float) > ws_size) return;      // insufficient scratch

    // 1) squared norms of x0/x1 rows
    {
        int rows = B_ * L_;
        dim3 g((rows + 7) / 8), blk(256);
        row_norm_kernel<<<g, blk, 0, stream>>>(x0, np_, rows, D0_);
        row_norm_kernel<<<g, blk, 0, stream>>>(x1, nq_, rows, D0_);
    }

    // 2) attention matrix A (gram GEMM + transform)
    attn_kernel<<<dim3(L_ / 16, 4, B_), 256, 0, stream>>>(x0, x1, np_, nq_, A);

    // 3) a0 = A @ W0, a1 = A^T @ W1
    {
        dim3 g(4, (D0_ + 15) / 16, B_);
        aw_kernel<false><<<g, 256, 0, stream>>>(A, W0, a0);
        aw_kernel<true ><<<g, 256, 0, stream>>>(A, W1, a1);
    }

    // 4-5) conv(f0) -> c0 ; conv(f1) -> c1  (M buffer reused sequentially)
    {
        dim3 gm(4, K_ / 16, B_);
        int totalC = B_ * LTP_ * K_;
        dim3 gt((totalC + 255) / 256);
        convm_kernel<<<gm, 256, 0, stream>>>(x0, a0, cw, M);
        tanhcomb_kernel<<<gt, 256, 0, stream>>>(M, cb, c0);
        convm_kernel<<<gm, 256, 0, stream>>>(x1, a1, cw, M);
        tanhcomb_kernel<<<gt, 256, 0, stream>>>(M, cb, c1);
    }

    // 6) norms of c0/c1 rows (pad rows are zero -> norm 0)
    {
        int rows = B_ * LTP_;
        dim3 g((rows + 7) / 8), blk(256);
        row_norm_kernel<<<g, blk, 0, stream>>>(c0, nc0, rows, K_);
        row_norm_kernel<<<g, blk, 0, stream>>>(c1, nc1, rows, K_);
    }

    // 7) A2 tile sums -> partial planes -> row/col vectors
    a2sum_kernel<<<dim3(LT_, (LT_ + 7) / 8, B_), 256, 0, stream>>>(
        c0, c1, nc0, nc1, rpart, cpart);
    {
        dim3 g((B_ * LP_ + 255) / 256);
        reduce_parts_kernel<<<g, 256, 0, stream>>>(rpart, rowv);
        reduce_parts_kernel<<<g, 256, 0, stream>>>(cpart, colv);
    }

    // 8) outputs: wp0 | wp1 | ap0 | ap1 (flat, in reference return order)
    {
        size_t wp_elems = (size_t)B_ * L_ * K_;
        dim3 g((unsigned)((wp_elems + 255) / 256));
        wp_kernel<<<g, 256, 0, stream>>>(c0, rowv, out);
        wp_kernel<<<g, 256, 0, stream>>>(c1, colv, out + wp_elems);
        ap_kernel<<<dim3(B_), K_, 0, stream>>>(c0, out + 2 * wp_elems);
        ap_kernel<<<dim3(B_), K_, 0, stream>>>(c1, out + 2 * wp_elems + B_ * K_);
    }
}